// GMM_55448027791416
// MI455X (gfx1250) — compile-verified
//
#include <hip/hip_runtime.h>
#include <math.h>

typedef float v2f __attribute__((ext_vector_type(2)));
typedef float v8f __attribute__((ext_vector_type(8)));

#define KK 256      // mixture components (output rows)
#define DD 64       // feature dim (WMMA reduction dim)
#define NN 262144   // samples (output cols)
#define NT 8        // 16-wide n-tiles per wave
#define KT 16       // K / 16 k-tiles

// ---------------------------------------------------------------------------
// Pre-kernel: per-component constants into workspace.
//   ws[k]      = c0[k]   = alpha[k] - 32*logvar[k] - 0.5*||mu_k||^2 * e^{-logvar[k]}
//   ws[256+k]  = invb[k] = e^{-logvar[k]}
// ---------------------------------------------------------------------------
__global__ void gmm_precomp(const float* __restrict__ mu,
                            const float* __restrict__ logvar,
                            const float* __restrict__ alpha,
                            float* __restrict__ ws) {
  const int k = threadIdx.x;  // one block of 256 threads, k in [0,256)
  float m2 = 0.0f;
  const float4* row = (const float4*)(mu + (size_t)k * DD);
#pragma unroll
  for (int j = 0; j < DD / 4; ++j) {
    float4 v = row[j];
    m2 += v.x * v.x + v.y * v.y + v.z * v.z + v.w * v.w;
  }
  const float lv   = logvar[k];
  const float invb = expf(-lv);
  ws[k]      = alpha[k] - 32.0f * lv - 0.5f * m2 * invb;
  ws[KK + k] = invb;
}

// ---------------------------------------------------------------------------
// Main kernel: cross = mu @ X^T via V_WMMA_F32_16X16X4_F32, fused epilogue,
// non-temporal streaming stores of the 256 MB output (preserve L2 for X).
// ---------------------------------------------------------------------------
__global__ void __launch_bounds__(256, 1)
gmm_main(const float* __restrict__ X,
         const float* __restrict__ mu,
         const float* __restrict__ ws,
         float* __restrict__ out) {
  const int lane  = threadIdx.x & 31;
  const int wave  = threadIdx.x >> 5;
  const int gw    = blockIdx.x * 8 + wave;   // global wave id
  const int ktile = gw & (KT - 1);           // fastest-varying -> X stays L2 hot
  const int ngrp  = gw >> 4;                 // which group of NT n-tiles
  const int idx   = lane & 15;
  const int half  = lane >> 4;
  const int k0    = ktile * 16;
  const int row0  = k0 + half * 8;           // C/D VGPR v maps to row row0+v

  // Per-row epilogue constants (uniform within each 16-lane half).
  float c0v[8], ibv[8];
#pragma unroll
  for (int v = 0; v < 8; ++v) {
    c0v[v] = ws[row0 + v];
    ibv[v] = ws[KK + row0 + v];
  }

  // A operand (mu k-tile) resident in registers for the wave's lifetime.
  // Lane (idx,half), step j holds mu[k0+idx][4j + 2*half .. +1]  (ISA 16x4 A layout).
  v2f A[16];
  {
    const float* mrow = mu + (size_t)(k0 + idx) * DD + 2 * half;
#pragma unroll
    for (int j = 0; j < 16; ++j) A[j] = *(const v2f*)(mrow + 4 * j);
  }

  const size_t n0base = (size_t)ngrp * (16 * NT);

  for (int t = 0; t < NT; ++t) {
    const size_t n0   = n0base + (size_t)t * 16;
    const float* xrow = X + (n0 + idx) * DD + 2 * half;

    // Prefetch next n-tile's X rows (global_prefetch_b8).
    if (t + 1 < NT) __builtin_prefetch(xrow + 16 * DD, 0, 1);

    // B operand: X n-tile, same per-lane float2 layout as A (4x16 B layout).
    v2f B[16];
#pragma unroll
    for (int j = 0; j < 16; ++j) B[j] = *(const v2f*)(xrow + 4 * j);

    // ||x_n||^2: each lane holds half the row; combine across lane halves.
    float x2p = 0.0f;
#pragma unroll
    for (int j = 0; j < 16; ++j) x2p += B[j].x * B[j].x + B[j].y * B[j].y;
    const float x2 = x2p + __shfl_xor(x2p, 16, 32);

    // cross[k][n] over D=64 via 16 chained f32 WMMAs.
    v8f acc = {};
#pragma unroll
    for (int j = 0; j < 16; ++j) {
      acc = __builtin_amdgcn_wmma_f32_16x16x4_f32(
          /*neg_a=*/false, A[j], /*neg_b=*/false, B[j],
          /*c_mod=*/(short)0, acc, /*reuse_a=*/false, /*reuse_b=*/false);
    }

    // Fused epilogue + streaming stores (row row0+v, col n0+idx).
#pragma unroll
    for (int v = 0; v < 8; ++v) {
      const float o = c0v[v] + ibv[v] * (acc[v] - 0.5f * x2);
      __builtin_nontemporal_store(o, out + (size_t)(row0 + v) * NN + n0 + idx);
    }
  }
}

extern "C" void kernel_launch(void* const* d_in, const int* in_sizes, int n_in,
                              void* d_out, int out_size, void* d_ws, size_t ws_size,
                              hipStream_t stream) {
  const float* X      = (const float*)d_in[0];
  const float* mu     = (const float*)d_in[1];
  const float* logvar = (const float*)d_in[2];
  const float* alpha  = (const float*)d_in[3];
  float*       ws     = (float*)d_ws;   // needs 2*256 floats = 2 KB
  float*       out    = (float*)d_out;

  gmm_precomp<<<1, 256, 0, stream>>>(mu, logvar, alpha, ws);

  // waves = KT * (NN/16/NT) = 16 * 2048 = 32768 -> 4096 blocks of 8 waves.
  const int blocks = (KT * (NN / 16 / NT)) / 8;
  gmm_main<<<blocks, 256, 0, stream>>>(X, mu, ws, out);
}